// QueryAndGroup_83528523972904
// MI455X (gfx1250) — compile-verified
//
#include <hip/hip_runtime.h>
#include <hip/hip_bf16.h>

typedef __attribute__((ext_vector_type(2))) float v2f;
typedef __attribute__((ext_vector_type(8))) float v8f;

#define BB 2
#define NN 16384
#define SS 4096
#define FC 64          // feature channels
#define NS 32          // nsample
#define R2 0.01f       // radius^2

#define HALFN (NN / 2)   // each wave scans half the point cloud
#define CHUNK 1024       // points staged into LDS per wave per iteration

__device__ __forceinline__ unsigned ballot32(bool p) {
  return __builtin_amdgcn_ballot_w32(p);
}

// ---------------------------------------------------------------------------
// Kernel A: ball query via V_WMMA_F32_16X16X4_F32.
//   A row m  = (cx, cy, cz, 1)
//   B col n  = (-2px, -2py, -2pz, |p|^2)
//   C[m]     = |c|^2
//   D = |c - p|^2 exactly, 256 distances per WMMA.
// Two waves per block share 16 centers; wave w scans points
// [w*8192, (w+1)*8192) into its own LDS buffer (no intra-scan barriers, free
// early exit), records ordered hits in LDS, then wave 0 merges (wave-0 hits
// all precede wave-1 hits in point order by construction).
// ---------------------------------------------------------------------------
__global__ __launch_bounds__(64) void ball_query_wmma_kernel(
    const float* __restrict__ xyz, const float* __restrict__ new_xyz,
    int* __restrict__ idx_out)
{
  // shp[w][j][0] = {-2x, -2y},  shp[w][j][1] = {-2z, |p|^2}
  __shared__ v2f shp[2][CHUNK][2];          // 32 KB
  __shared__ int sh_hits[2][16][NS];        // 4 KB
  __shared__ int sh_cnt[2][16];

  const int blocks_per_batch = SS / 16;     // 256
  const int b      = blockIdx.x / blocks_per_batch;
  const int s_base = (blockIdx.x % blocks_per_batch) * 16;
  const int w    = threadIdx.x >> 5;        // wave: which half of N
  const int lane = threadIdx.x & 31;
  const int half = lane >> 4;               // 0: K0/K1 half, 1: K2/K3 half
  const int lm   = lane & 15;

  const int sc = s_base + lm;               // this lane's center
  const float* cptr = new_xyz + ((size_t)b * SS + sc) * 3;
  const float cx = cptr[0], cy = cptr[1], cz = cptr[2];
  const float cc = cx * cx + cy * cy + cz * cz;

  // A matrix (16x4 f32): vgpr0 = K0|K2, vgpr1 = K1|K3 per lane-half.
  v2f A;
  A.x = half ? cz : cx;
  A.y = half ? 1.0f : cy;

  // C matrix: Cm[r] = |c|^2 of center M = r + 8*half.
  v8f Cm;
#pragma unroll
  for (int r = 0; r < 8; ++r) {
    Cm[r] = __shfl(cc, r + (half ? 8 : 0), 32);
  }

  int cnt = 0;
  bool done = false;

  const int scan_lo = w * HALFN;
  const int scan_hi = scan_lo + HALFN;

  for (int cbase = scan_lo; cbase < scan_hi && !done; cbase += CHUNK) {
    // Stage this wave's chunk into its own LDS buffer (wave-private).
    for (int i = lane; i < CHUNK; i += 32) {
      const float* q = xyz + ((size_t)b * NN + cbase + i) * 3;
      const float x = q[0], y = q[1], z = q[2];
      v2f t0, t1;
      t0.x = -2.0f * x;            t0.y = -2.0f * y;
      t1.x = -2.0f * z;            t1.y = x * x + y * y + z * z;
      shp[w][i][0] = t0;
      shp[w][i][1] = t1;
    }
    // Per-wave DS ordering is sufficient; wait drains the stores.
    asm volatile("s_wait_dscnt 0" ::: "memory");

    // Software-pipelined tile loop: prefetch B for tile t+1 before WMMA(t).
    v2f Bv = shp[w][lm][half];
    for (int t = 0; t < CHUNK; t += 16) {
      const int jn = (t + 16 < CHUNK) ? (t + 16 + lm) : lm;
      v2f Bn = shp[w][jn][half];

      // 8 args: (neg_a, A, neg_b, B, c_mod, C, reuse_a, reuse_b)
      v8f d2 = __builtin_amdgcn_wmma_f32_16x16x4_f32(
          false, A, false, Bv, (short)0, Cm, false, false);

      unsigned bal[8];
#pragma unroll
      for (int r = 0; r < 8; ++r) bal[r] = ballot32(d2[r] < R2);

      // Wave-uniform any-hit test: skip all mask processing on empty tiles.
      const unsigned any = (bal[0] | bal[1]) | (bal[2] | bal[3]) |
                           ((bal[4] | bal[5]) | (bal[6] | bal[7]));
      if (any) {
        // Owner lane l (<16) assembles the ordered 16-bit hit mask of its
        // center: M=l<8 -> low half of bal[l]; M=l>=8 -> high half of bal[l-8].
        unsigned m = 0;
#pragma unroll
        for (int r = 0; r < 8; ++r) {
          m = (lane == r)     ? (bal[r] & 0xFFFFu) : m;
          m = (lane == r + 8) ? (bal[r] >> 16)     : m;
        }
        if (lane < 16) {
          while (m && cnt < NS) {
            const int bit = __builtin_ctz(m);
            m &= m - 1;
            sh_hits[w][lane][cnt] = cbase + t + bit;  // ascending point order
            ++cnt;
          }
        }
        const unsigned db = ballot32(cnt >= NS);
        if ((db & 0xFFFFu) == 0xFFFFu) { done = true; break; }
      }
      Bv = Bn;
    }
  }

  if (lane < 16) sh_cnt[w][lane] = cnt;
  __syncthreads();

  // Merge: wave-0 hits all have smaller indices than wave-1 hits, so the
  // first NS of the concatenation is exactly the reference selection.
  if (w == 0) {
    const int m = lane & 15;
    const int h = lane >> 4;
    const int c0 = sh_cnt[0][m];
    const int c1 = sh_cnt[1][m];
    int first = 0;
    if (c0 > 0)      first = sh_hits[0][m][0];
    else if (c1 > 0) first = sh_hits[1][m][0];
    const int base = (b * SS + s_base + m) * NS;
    for (int k = h * 16; k < h * 16 + 16; ++k) {
      int v;
      if (k < c0)            v = sh_hits[0][m][k];
      else if (k - c0 < c1)  v = sh_hits[1][m][k - c0];
      else                   v = first;   // pad (0 when no hits at all)
      idx_out[base + k] = v;
    }
  }
}

// ---------------------------------------------------------------------------
// Kernel B: gather + write (B, 3+FC, S, NS). One wave per center, lane = k.
// Stores are 128B-coalesced per (channel, center); feature gathers hit L2
// (features = 8.4 MB << 192 MB L2).
// ---------------------------------------------------------------------------
__global__ __launch_bounds__(256) void group_kernel(
    const float* __restrict__ xyz, const float* __restrict__ new_xyz,
    const float* __restrict__ features, const int* __restrict__ idx,
    float* __restrict__ out)
{
  const int wave = threadIdx.x >> 5;
  const int lane = threadIdx.x & 31;
  const int g = blockIdx.x * 8 + wave;     // global center id in [0, B*S)
  const int b = g / SS;
  const int s = g % SS;

  const int id = idx[(size_t)g * NS + lane];
  const float* p = xyz + ((size_t)b * NN + id) * 3;
  const float px = p[0], py = p[1], pz = p[2];
  const float* c = new_xyz + (size_t)g * 3;

  const size_t chs   = (size_t)SS * NS;                       // channel stride
  const size_t sbase = (size_t)b * (3 + FC) * chs + (size_t)s * NS + lane;

  out[sbase + 0 * chs] = px - c[0];
  out[sbase + 1 * chs] = py - c[1];
  out[sbase + 2 * chs] = pz - c[2];

  const float* frow = features + (size_t)b * FC * NN + id;
#pragma unroll 4
  for (int ch = 0; ch < FC; ++ch) {
    out[sbase + (size_t)(3 + ch) * chs] = frow[(size_t)ch * NN];
  }
}

extern "C" void kernel_launch(void* const* d_in, const int* in_sizes, int n_in,
                              void* d_out, int out_size, void* d_ws, size_t ws_size,
                              hipStream_t stream) {
  (void)in_sizes; (void)n_in; (void)out_size; (void)ws_size;
  const float* xyz      = (const float*)d_in[0];
  const float* new_xyz  = (const float*)d_in[1];
  const float* features = (const float*)d_in[2];
  float* out = (float*)d_out;
  int* idx_ws = (int*)d_ws;   // B*S*NS ints = 1 MB

  dim3 gridA(BB * (SS / 16));               // 512 blocks, 2 waves each
  ball_query_wmma_kernel<<<gridA, 64, 0, stream>>>(xyz, new_xyz, idx_ws);

  dim3 gridB(BB * SS / 8);                  // 1024 blocks
  group_kernel<<<gridB, 256, 0, stream>>>(xyz, new_xyz, features, idx_ws, out);
}